// RoiAlign_4372276707982
// MI455X (gfx1250) — compile-verified
//
#include <hip/hip_runtime.h>
#include <hip/hip_bf16.h>

// ROI-Align for MI455X (gfx1250).
//
// Bandwidth-bound gather workload (~500 MB traffic, ~0.2 GFLOP reference
// math). We exploit separability of bilinear crop-and-resize to express the
// interpolation as two tiny GEMMs per box:
//   stage1: xtmp[iy, px, c] = sum_ix Wx[px,ix] * F[yrow[iy], xcol[ix], c]
//   stage2: out [py, px, c] = sum_iy Wy[py,iy] * xtmp[iy, px, c]
// both executed with V_WMMA_F32_16X16X4_F32 (M=16 pixels, K=4-chunks of the
// 14 padded taps, N=16-channel chunks). Gathers load straight into the WMMA
// B-fragment register layout with 64B-per-16-lane coalescing.

typedef __attribute__((ext_vector_type(2))) float v2f;
typedef __attribute__((ext_vector_type(8))) float v8f;

#define POOL 7
#define NTAP 14              // 7 * {lo, hi} taps per axis (padded to K=16)
#define CCH  256             // channels
#define CHALF 128            // channels processed per half (LDS budget)

__device__ __forceinline__ int iminc(int a, int b) { return a < b ? a : b; }
__device__ __forceinline__ int imaxc(int a, int b) { return a > b ? a : b; }

// One 1-D sample: s = base + p*step; corner indices (clipped), frac, validity.
__device__ __forceinline__ void sample1d(float base, float step, int p, int L,
                                         int& i0, int& i1, float& fr, bool& valid) {
    float s  = base + step * (float)p;
    float s0 = floorf(s);
    fr = s - s0;
    int i = (int)s0;
    i0 = iminc(imaxc(i, 0), L - 1);
    i1 = iminc(i0 + 1, L - 1);
    valid = (s >= 0.0f) && (s <= (float)(L - 1));
}

__global__ __launch_bounds__(256) void roi_align_wmma(
    const float* __restrict__ boxes,     // [B,N,4] (y1,x1,y2,x2)
    const int*   __restrict__ img_shape, // [3]
    const float* __restrict__ f2,        // [B,256,256,C]
    const float* __restrict__ f3,        // [B,128,128,C]
    const float* __restrict__ f4,        // [B, 64, 64,C]
    const float* __restrict__ f5,        // [B, 32, 32,C]
    float*       __restrict__ out,       // [B,N,7,7,C]
    int B, int N)
{
    // xtmp[iy][px][c-half] : stage-1 output, stage-2 input (transpose medium)
    __shared__ float xtmp[NTAP * POOL * CHALF];   // 50176 B

    const int m     = blockIdx.x;          // global box id
    const int tid   = threadIdx.x;
    const int wave  = tid >> 5;            // 8 waves / block (wave32)
    const int lane  = tid & 31;
    const int hh    = lane >> 4;           // half-wave: 0 -> lanes 0-15, 1 -> 16-31
    const int l15   = lane & 15;

    // ---- per-box geometry (uniform across the block) -----------------------
    const float y1 = boxes[m * 4 + 0];
    const float x1 = boxes[m * 4 + 1];
    const float y2 = boxes[m * 4 + 2];
    const float x2 = boxes[m * 4 + 3];
    const float bh = y2 - y1, bw = x2 - x1;

    const float area = (float)img_shape[0] * (float)img_shape[1];
    // lvl = clip(4 + round(log2(sqrt(w*h)*sqrt(area)/224)), 2, 5); rintf == round-half-even
    float lvlf = 4.0f + rintf(log2f(sqrtf(bw * bh) * sqrtf(area) * (1.0f / 224.0f)));
    lvlf = fminf(fmaxf(lvlf, 2.0f), 5.0f);
    const int lvl = (int)lvlf;

    const float* feat = (lvl == 2) ? f2 : (lvl == 3) ? f3 : (lvl == 4) ? f4 : f5;
    const int H = 1024 >> lvl;             // 256/128/64/32
    const int W = H;
    const int b = m / N;
    const float* fbase = feat + (size_t)b * H * W * CCH;

    const float Hm1 = (float)(H - 1), Wm1 = (float)(W - 1);
    const float ystep = bh * Hm1 * (1.0f / (POOL - 1));
    const float xstep = bw * Wm1 * (1.0f / (POOL - 1));
    const float ybase = y1 * Hm1;
    const float xbase = x1 * Wm1;

    // ---- per-lane WMMA A fragments (constant for the whole box) ------------
    // Stage-1 A = Wx[px, ix] at (m = l15, k = v + 2*hh within each K=4 step).
    int xi0, xi1; float fx; bool vx;
    sample1d(xbase, xstep, l15, W, xi0, xi1, fx, vx);
    v2f A1[4];
#pragma unroll
    for (int s = 0; s < 4; ++s) {
#pragma unroll
        for (int v = 0; v < 2; ++v) {
            int ix = 4 * s + v + 2 * hh;
            float w = 0.0f;
            if (l15 < POOL && vx) {
                if (ix < POOL)            w = (ix == l15)        ? (1.0f - fx) : 0.0f;
                else if (ix < 2 * POOL)   w = ((ix - POOL) == l15) ? fx        : 0.0f;
            }
            if (v == 0) A1[s].x = w; else A1[s].y = w;
        }
    }

    // Stage-2 A = Wy[py, iy], same fragment layout.
    int yi0d, yi1d; float fy; bool vy;
    sample1d(ybase, ystep, l15, H, yi0d, yi1d, fy, vy);
    v2f A2[4];
#pragma unroll
    for (int s = 0; s < 4; ++s) {
#pragma unroll
        for (int v = 0; v < 2; ++v) {
            int iy = 4 * s + v + 2 * hh;
            float w = 0.0f;
            if (l15 < POOL && vy) {
                if (iy < POOL)            w = (iy == l15)        ? (1.0f - fy) : 0.0f;
                else if (iy < 2 * POOL)   w = ((iy - POOL) == l15) ? fy        : 0.0f;
            }
            if (v == 0) A2[s].x = w; else A2[s].y = w;
        }
    }

    // Stage-1 B gather columns: xcol(ix) for this lane's (s,v) slots.
    int cols[4][2];
#pragma unroll
    for (int s = 0; s < 4; ++s) {
#pragma unroll
        for (int v = 0; v < 2; ++v) {
            int ix = 4 * s + v + 2 * hh;
            int p  = (ix < POOL) ? ix : ((ix < 2 * POOL) ? ix - POOL : 0);
            int c0, c1; float fr; bool vv;
            sample1d(xbase, xstep, p, W, c0, c1, fr, vv);
            cols[s][v] = (ix < POOL) ? c0 : ((ix < 2 * POOL) ? c1 : c0);
        }
    }

    const v8f vzero = {0.f, 0.f, 0.f, 0.f, 0.f, 0.f, 0.f, 0.f};

    // ---- two channel halves (LDS reuse) ------------------------------------
    for (int hhalf = 0; hhalf < 2; ++hhalf) {
        const int cofs = hhalf * CHALF;

        // ======== stage 1: 14 iy-rows x 8 c-chunks, split over 8 waves ======
#pragma unroll 1
        for (int g = wave * 14; g < wave * 14 + 14; ++g) {
            const int iy = g >> 3;     // 0..13
            const int cc = g & 7;      // 0..7
            // yrow (uniform per group): iy<7 -> y0[iy], else y1[iy-7]
            {
                const int p = (iy < POOL) ? iy : iy - POOL;
                int r0, r1; float fr; bool vv;
                sample1d(ybase, ystep, p, H, r0, r1, fr, vv);
                const int yr = (iy < POOL) ? r0 : r1;

                const float* rb = fbase + (size_t)yr * W * CCH + (cofs + cc * 16 + l15);
                v8f acc = vzero;
#pragma unroll
                for (int s = 0; s < 4; ++s) {
                    v2f Bf;
                    Bf.x = rb[(size_t)cols[s][0] * CCH];   // coalesced: 64B / 16 lanes
                    Bf.y = rb[(size_t)cols[s][1] * CCH];
                    acc = __builtin_amdgcn_wmma_f32_16x16x4_f32(
                        false, A1[s], false, Bf, (short)0, acc, false, false);
                }
                // D row r holds px = r + 8*hh; only px<7 (hh==0, r<7) is real.
                if (hh == 0) {
#pragma unroll
                    for (int r = 0; r < POOL; ++r)
                        xtmp[(iy * POOL + r) * CHALF + cc * 16 + l15] = acc[r];
                }
            }
        }
        __syncthreads();

        // ======== stage 2: 7 px x 8 c-chunks, split over 8 waves ============
#pragma unroll 1
        for (int g = wave * 7; g < wave * 7 + 7; ++g) {
            const int px = g >> 3;     // 0..6
            const int cc = g & 7;      // 0..7
            v8f acc = vzero;
#pragma unroll
            for (int s = 0; s < 4; ++s) {
                v2f Bf;
                int k0 = iminc(4 * s + 0 + 2 * hh, NTAP - 1);  // A2 is 0 beyond 13
                int k1 = iminc(4 * s + 1 + 2 * hh, NTAP - 1);
                Bf.x = xtmp[(k0 * POOL + px) * CHALF + cc * 16 + l15];
                Bf.y = xtmp[(k1 * POOL + px) * CHALF + cc * 16 + l15];
                acc = __builtin_amdgcn_wmma_f32_16x16x4_f32(
                    false, A2[s], false, Bf, (short)0, acc, false, false);
            }
            if (hh == 0) {
                float* op = out + ((size_t)m * (POOL * POOL)) * CCH
                                + (size_t)px * CCH + cofs + cc * 16 + l15;
#pragma unroll
                for (int r = 0; r < POOL; ++r)     // py = r
                    op[(size_t)r * POOL * CCH] = acc[r];
            }
        }
        __syncthreads();   // protect xtmp before next half overwrites it
    }
}

extern "C" void kernel_launch(void* const* d_in, const int* in_sizes, int n_in,
                              void* d_out, int out_size, void* d_ws, size_t ws_size,
                              hipStream_t stream) {
    const float* boxes = (const float*)d_in[0];
    const int*   img   = (const int*)d_in[1];
    const float* f2    = (const float*)d_in[2];
    const float* f3    = (const float*)d_in[3];
    const float* f4    = (const float*)d_in[4];
    const float* f5    = (const float*)d_in[5];

    const int BN = in_sizes[0] / 4;                 // B*N boxes
    int B = in_sizes[2] / (256 * 256 * 256);        // feat2 = [B,256,256,256]
    if (B < 1) B = 1;
    const int N = BN / B;

    roi_align_wmma<<<BN, 256, 0, stream>>>(boxes, img, f2, f3, f4, f5,
                                           (float*)d_out, B, N);
}